// CropRandomizer_6459630813207
// MI455X (gfx1250) — compile-verified
//
#include <hip/hip_runtime.h>

// CropRandomizer forward (pos_enc=True) for MI455X / gfx1250.
//
// Pure bandwidth op: 478 MB f32 output stream, ~287 MB input reads (input is
// 44 MB, fits in 192 MB L2 and is re-read ~8x by overlapping crops).
// Strategy: 1 thread = 1 aligned float4 of output.
//   - image channels (c<3): 4x dword loads (w0 is only 4B aligned), L2-resident
//   - pos channels (c==3,4): computed, zero bytes read
//   - output written with non-temporal 128-bit stores (th:TH_STORE_NT) so the
//     478 MB stream does not evict the hot input from L2.
// WMMA is inapplicable (zero matrix FLOPs); the gfx1250-specific win here is
// the NT cache-control path on wide stores.

#define B_IMGS    64
#define C_IMG     3
#define CP        5          // C + 2 pos channels
#define H_IMG     240
#define W_IMG     240
#define CROP_H    216
#define CROP_W    216
#define NUM_CROPS 8
#define X4_ROW    (CROP_W / 4)                 // 54 float4 per output row
#define TOTAL4    (B_IMGS * NUM_CROPS * CP * CROP_H * X4_ROW)  // 29,859,840

typedef float float4v __attribute__((ext_vector_type(4)));

__global__ __launch_bounds__(256)
void crop_pos_nt_kernel(const float* __restrict__ images,
                        const int*   __restrict__ crop_h,
                        const int*   __restrict__ crop_w,
                        float*       __restrict__ out)
{
    unsigned idx = blockIdx.x * 256u + threadIdx.x;   // float4 index into out
    if (idx >= (unsigned)TOTAL4) return;

    // Decompose flat float4 index: [n=512][c=5][y=216][x4=54]
    unsigned x4 = idx % X4_ROW;
    unsigned t  = idx / X4_ROW;
    unsigned y  = t % CROP_H;
    t          /= CROP_H;
    unsigned c  = t % CP;
    unsigned n  = t / CP;                 // 0..511  == b*NUM_CROPS + crop

    // crop_inds_h / crop_inds_w are [B][NUM_CROPS] int32, flat index == n.
    int h0 = crop_h[n];
    int w0 = crop_w[n];

    unsigned x0 = x4 * 4u;               // first of 4 output columns
    float4v v;

    if (c < C_IMG) {
        // Source row: images[b][c][h0+y][w0+x0 .. +3]; 4B-aligned only.
        unsigned b   = n / NUM_CROPS;
        unsigned off = ((b * C_IMG + c) * (unsigned)H_IMG + (unsigned)(h0 + (int)y))
                         * (unsigned)W_IMG
                       + (unsigned)(w0 + (int)x0);
        const float* src = images + off;
        v.x = src[0];
        v.y = src[1];
        v.z = src[2];
        v.w = src[3];
    } else if (c == C_IMG) {
        // pos channel 0: normalized source row coordinate (uniform across x)
        float pv = (float)(h0 + (int)y) / (float)H_IMG;
        v.x = pv; v.y = pv; v.z = pv; v.w = pv;
    } else {
        // pos channel 1: normalized source column coordinate
        float bx = (float)(w0 + (int)x0);
        v.x = bx            / (float)W_IMG;
        v.y = (bx + 1.0f)   / (float)W_IMG;
        v.z = (bx + 2.0f)   / (float)W_IMG;
        v.w = (bx + 3.0f)   / (float)W_IMG;
    }

    // Output layout is exactly [n][c][y][x] row-major, so the float4 slot is
    // just `idx`. Base from hipMalloc is 256B aligned; idx*16 keeps 16B align.
    // Non-temporal: stream 478 MB past L2, keep the 44 MB input resident.
    __builtin_nontemporal_store(v, (float4v*)out + idx);
}

extern "C" void kernel_launch(void* const* d_in, const int* in_sizes, int n_in,
                              void* d_out, int out_size, void* d_ws, size_t ws_size,
                              hipStream_t stream)
{
    const float* images = (const float*)d_in[0];
    const int*   ch     = (const int*)  d_in[1];
    const int*   cw     = (const int*)  d_in[2];
    float*       out    = (float*)      d_out;

    // TOTAL4 = 29,859,840 = 116,640 * 256 exactly.
    dim3 grid((TOTAL4 + 255u) / 256u);
    dim3 block(256);
    hipLaunchKernelGGL(crop_pos_nt_kernel, grid, block, 0, stream,
                       images, ch, cw, out);
}